// SLEDLoss_197568496220
// MI455X (gfx1250) — compile-verified
//
#include <hip/hip_runtime.h>
#include <math.h>

typedef __attribute__((ext_vector_type(2))) float v2f;
typedef __attribute__((ext_vector_type(8))) float v8f;

#define C_DIM   301
#define EMPTY_C 300
#define S_SLOTS 8

// One block (256 threads = 8 wave32) per (b,t) frame.
__global__ __launch_bounds__(256) void sled_frame_kernel(
    const float* __restrict__ class_logits,   // (F,8,301)
    const float* __restrict__ doa_vec,        // (F,8,3)
    const float* __restrict__ loudness,       // (F,8)
    const float* __restrict__ confidence,     // (F,8)
    const float* __restrict__ sce_vec,        // (F,8,3)
    const int*   __restrict__ gt_cls,         // (F,8)
    const float* __restrict__ gt_doa,         // (F,8,3)
    const float* __restrict__ gt_loud,        // (F,8)
    const unsigned char* __restrict__ gt_mask,// (F,8) bool
    float* __restrict__ ws)                   // (F,5) per-frame partials
{
  __shared__ float smax[8], ssum[8];
  __shared__ float dnA[8][3];      // normalized pred DOAs
  __shared__ float gdnR[8][3];     // normalized gt DOAs (all slots)
  __shared__ float cpL[8];         // sigmoid(conf)
  __shared__ int   actL[8], gcL[8], Gsh;
  __shared__ float cost[8][8];
  __shared__ int   matchedL[8], tgtL[8], gidxL[8];
  __shared__ float fL[8], dL[8], slL[8], bceL[8], sqL[8], mfL[8];

  const int tid  = threadIdx.x;
  const int wave = tid >> 5;
  const int lane = tid & 31;
  const int frame = blockIdx.x;
  const size_t fbase = (size_t)frame * S_SLOTS;

  // ---- Phase 0: per-slot online softmax stats (wave w -> slot w) ----
  {
    const float* cl = class_logits + (fbase + wave) * C_DIM;
    float m = -3.402823e38f;
    float s = 0.f;
    for (int c = lane; c < C_DIM; c += 32) {        // coalesced, single pass
      float x = cl[c];
      if (x > m) { s *= expf(m - x); m = x; }
      s += expf(x - m);
    }
    for (int off = 16; off > 0; off >>= 1) {        // wave32 combine
      float m2 = __shfl_xor(m, off, 32);
      float s2 = __shfl_xor(s, off, 32);
      float nm = fmaxf(m, m2);
      s = s * expf(m - nm) + s2 * expf(m2 - nm);
      m = nm;
    }
    if (lane == 0) { smax[wave] = m; ssum[wave] = s; }
  }

  // ---- small per-slot data ----
  if (tid < 8) {
    int s = tid;
    float x = doa_vec[(fbase + s) * 3 + 0];
    float y = doa_vec[(fbase + s) * 3 + 1];
    float z = doa_vec[(fbase + s) * 3 + 2];
    float n = fmaxf(sqrtf(x * x + y * y + z * z), 1e-12f);
    dnA[s][0] = x / n; dnA[s][1] = y / n; dnA[s][2] = z / n;
    float cf = confidence[fbase + s];
    cpL[s] = 1.f / (1.f + expf(-cf));
    matchedL[s] = 0; tgtL[s] = EMPTY_C; gidxL[s] = 0;
  } else if (tid < 16) {
    int s = tid - 8;
    float x = gt_doa[(fbase + s) * 3 + 0];
    float y = gt_doa[(fbase + s) * 3 + 1];
    float z = gt_doa[(fbase + s) * 3 + 2];
    float n = fmaxf(sqrtf(x * x + y * y + z * z), 1e-12f);
    gdnR[s][0] = x / n; gdnR[s][1] = y / n; gdnR[s][2] = z / n;
  }
  __syncthreads();

  if (tid == 0) {                       // compact active gt slots
    int g = 0;
    for (int s = 0; s < 8; s++)
      if (gt_mask[fbase + s]) { actL[g] = s; gcL[g] = gt_cls[fbase + s]; g++; }
    Gsh = g;
  }
  __syncthreads();
  const int G = Gsh;

  // ---- cost base: -prob[p][gc_j] + L_CONF*(1-cp_p)  (threads 0..63) ----
  if (tid < 64) {
    int p = tid >> 3, j = tid & 7;
    if (j < G) {
      float x = class_logits[(fbase + p) * C_DIM + gcL[j]];   // L2-resident gather
      float prob = expf(x - smax[p]) / ssum[p];
      cost[p][j] = -prob + 0.5f * (1.f - cpL[p]);
    }
  }
  __syncthreads();

  // ---- DOA Gram via V_WMMA_F32_16X16X4_F32 (wave 0, EXEC all-ones) ----
  if (wave == 0) {
    v2f a, b;
    v8f c = {};
    const int half = lane >> 4;
    const int idx  = lane & 15;
    if (half == 0) {                 // this half supplies K=0,1
      a.x = (idx < 8) ? dnA[idx][0] : 0.f;
      a.y = (idx < 8) ? dnA[idx][1] : 0.f;
      b.x = (idx < G) ? gdnR[actL[idx]][0] : 0.f;
      b.y = (idx < G) ? gdnR[actL[idx]][1] : 0.f;
    } else {                         // this half supplies K=2,3 (K=3 zero-padded)
      a.x = (idx < 8) ? dnA[idx][2] : 0.f;
      a.y = 0.f;
      b.x = (idx < G) ? gdnR[actL[idx]][2] : 0.f;
      b.y = 0.f;
    }
    v8f dacc = __builtin_amdgcn_wmma_f32_16x16x4_f32(
        false, a, false, b, (short)0, c, false, false);
    if (lane < 8 && lane < G) {      // lane j owns column j: D vgpr p = (M=p,N=j)
      int j = lane;
#pragma unroll
      for (int p = 0; p < 8; p++) cost[p][j] += 1.0f - dacc[p];
    }
  }
  __syncthreads();

  // ---- Hungarian (thread 0, float64, faithful to reference _lsa) ----
  if (tid == 0 && G > 0) {
    const bool transposed = (S_SLOTS > G);   // rows(P=8) > cols(G)
    const int n = transposed ? G : S_SLOTS;
    const int m = transposed ? S_SLOTS : G;
    double u[9], v[9];
    int pj[9], wayv[9];
    for (int k = 0; k <= m; k++) { v[k] = 0.0; pj[k] = 0; }
    for (int k = 0; k <= n; k++) u[k] = 0.0;
    for (int i = 1; i <= n; i++) {
      pj[0] = i;
      int j0 = 0;
      double minv[9]; bool used[9];
      for (int j = 0; j <= m; j++) { minv[j] = 1e300; used[j] = false; wayv[j] = 0; }
      while (true) {
        used[j0] = true;
        int i0 = pj[j0];
        double delta = 1e300;
        int j1 = 0;
        for (int j = 1; j <= m; j++) if (!used[j]) {
          double cij = transposed ? (double)cost[j - 1][i0 - 1]
                                  : (double)cost[i0 - 1][j - 1];
          double cur = cij - u[i0] - v[j];
          if (cur < minv[j]) { minv[j] = cur; wayv[j] = j0; }
          if (minv[j] < delta) { delta = minv[j]; j1 = j; }
        }
        for (int j = 0; j <= m; j++) {
          if (used[j]) { u[pj[j]] += delta; v[j] -= delta; }
          else minv[j] -= delta;
        }
        j0 = j1;
        if (pj[j0] == 0) break;
      }
      while (j0) { int j1 = wayv[j0]; pj[j0] = pj[j1]; j0 = j1; }
    }
    for (int j = 1; j <= m; j++) if (pj[j]) {
      int r = pj[j] - 1, cI = j - 1;
      int pred  = transposed ? cI : r;     // prediction slot
      int gslot = transposed ? r  : cI;    // order in active-gt list
      matchedL[pred] = 1;
      tgtL[pred]  = gcL[gslot];
      gidxL[pred] = actL[gslot];
    }
  }
  __syncthreads();

  // ---- per-slot loss terms (threads 0..7) ----
  if (tid < 8) {
    int s = tid;
    size_t sl = fbase + s;
    float mf = matchedL[s] ? 1.f : 0.f;
    // focal CE with target class (EMPTY if unmatched)
    float lse = smax[s] + logf(ssum[s]);
    float xt = class_logits[sl * C_DIM + tgtL[s]];
    float ce = lse - xt;
    float om = 1.f - expf(-ce);
    fL[s] = 0.25f * om * om * ce;            // ALPHA=0.25, GAMMA=2
    // cosine DOA vs matched gt
    float dx = doa_vec[sl * 3 + 0], dy = doa_vec[sl * 3 + 1], dz = doa_vec[sl * 3 + 2];
    size_t gl = fbase + gidxL[s];
    float mx = gt_doa[gl * 3 + 0], my = gt_doa[gl * 3 + 1], mz = gt_doa[gl * 3 + 2];
    float num = dx * mx + dy * my + dz * mz;
    float den = fmaxf(sqrtf(dx * dx + dy * dy + dz * dz), 1e-8f) *
                fmaxf(sqrtf(mx * mx + my * my + mz * mz), 1e-8f);
    dL[s] = 1.f - num / den;
    // smooth-L1 loudness
    float ml = gt_loud[gl];
    float diff = loudness[sl] - ml;
    float ad = fabsf(diff);
    slL[s] = (ad < 1.f) ? 0.5f * diff * diff : ad - 0.5f;
    // BCE confidence (stable softplus)
    float cf = confidence[sl];
    bceL[s] = fmaxf(cf, 0.f) + log1pf(expf(-fabsf(cf))) - cf * mf;
    // SCE MSE
    float esc = expf(ml / 20.f);
    float e0 = sce_vec[sl * 3 + 0] - mx * esc;
    float e1 = sce_vec[sl * 3 + 1] - my * esc;
    float e2 = sce_vec[sl * 3 + 2] - mz * esc;
    sqL[s] = (e0 * e0 + e1 * e1 + e2 * e2) * (1.f / 3.f);
    mfL[s] = mf;
  }
  __syncthreads();

  // ---- per-frame combine (thread 0) ----
  if (tid == 0) {
    float Gm = 0, sf = 0, sfu = 0, sd = 0, ssl = 0, sbce = 0, ssq = 0;
    for (int s = 0; s < 8; s++) {
      float mf = mfL[s];
      Gm += mf;
      sf += fL[s] * mf;  sfu += fL[s] * (1.f - mf);
      sd += dL[s] * mf;  ssl += slL[s] * mf;
      sbce += bceL[s];   ssq += sqL[s] * mf;
    }
    float Um = (float)S_SLOTS - Gm;
    float sG = fmaxf(Gm, 1.f), sU = fmaxf(Um, 1.f);
    float cls_fr  = ((Gm > 0.f) ? sf / sG : 0.f) + ((Um > 0.f) ? sfu / sU : 0.f);
    float doa_fr  = (Gm > 0.f) ? sd  / sG : 0.f;
    float loud_fr = (Gm > 0.f) ? ssl / sG : 0.f;
    float conf_fr = sbce * (1.f / (float)S_SLOTS);
    float sce_fr  = (Gm > 0.f) ? ssq / sG : 0.f;
    float* w = ws + (size_t)frame * 5;
    w[0] = cls_fr; w[1] = doa_fr; w[2] = loud_fr; w[3] = conf_fr; w[4] = sce_fr;
  }
}

// Deterministic final reduction: fixed per-thread stride order + fixed LDS tree.
__global__ __launch_bounds__(256) void sled_reduce_kernel(
    const float* __restrict__ ws, float* __restrict__ out, int frames)
{
  __shared__ float acc[5][256];
  int t = threadIdx.x;
  float a0 = 0, a1 = 0, a2 = 0, a3 = 0, a4 = 0;
  for (int f = t; f < frames; f += 256) {
    const float* w = ws + (size_t)f * 5;
    a0 += w[0]; a1 += w[1]; a2 += w[2]; a3 += w[3]; a4 += w[4];
  }
  acc[0][t] = a0; acc[1][t] = a1; acc[2][t] = a2; acc[3][t] = a3; acc[4][t] = a4;
  __syncthreads();
  for (int off = 128; off > 0; off >>= 1) {
    if (t < off) {
#pragma unroll
      for (int k = 0; k < 5; k++) acc[k][t] += acc[k][t + off];
    }
    __syncthreads();
  }
  if (t == 0) {
    const float inv = 1.f / (float)frames;             // norm = B*T
    float cls_l  = 1.0f * acc[0][0] * inv;             // W_CLS
    float doa_l  = 1.0f * acc[1][0] * inv;             // W_DOA
    float loud_l = 0.5f * acc[2][0] * inv;             // W_LOUD
    float conf_l = 1.0f * acc[3][0] * inv;             // W_CONF
    float sce_l  = 0.3f * acc[4][0] * inv;             // W_SCE
    out[0] = cls_l; out[1] = doa_l; out[2] = loud_l;
    out[3] = conf_l; out[4] = sce_l;
    out[5] = cls_l + doa_l + loud_l + conf_l + sce_l;
  }
}

extern "C" void kernel_launch(void* const* d_in, const int* in_sizes, int n_in,
                              void* d_out, int out_size, void* d_ws, size_t ws_size,
                              hipStream_t stream) {
  const float* class_logits = (const float*)d_in[0];
  const float* doa_vec      = (const float*)d_in[1];
  const float* loud         = (const float*)d_in[2];
  const float* conf         = (const float*)d_in[3];
  const float* sce_vec      = (const float*)d_in[4];
  const int*   gcls         = (const int*)  d_in[5];
  const float* gdoa         = (const float*)d_in[6];
  const float* gloud        = (const float*)d_in[7];
  const unsigned char* gmask = (const unsigned char*)d_in[8];
  float* out = (float*)d_out;
  float* ws  = (float*)d_ws;

  const int frames = in_sizes[2] / S_SLOTS;   // loudness is (B,T,S)

  sled_frame_kernel<<<frames, 256, 0, stream>>>(
      class_logits, doa_vec, loud, conf, sce_vec, gcls, gdoa, gloud, gmask, ws);
  sled_reduce_kernel<<<1, 256, 0, stream>>>(ws, out, frames);
}